// Fit_64175401337595
// MI455X (gfx1250) — compile-verified
//
#include <hip/hip_runtime.h>
#include <hip/hip_bf16.h>
#include <stdint.h>

#define TOPK_ 20
#define SELECTS_ 100

typedef __attribute__((ext_vector_type(2))) float v2f;
typedef __attribute__((ext_vector_type(8))) float v8f;

// ---------------------------------------------------------------------------
// helpers
// ---------------------------------------------------------------------------
__device__ inline float block_reduce_sum(float v, float* red) {
  int tid = threadIdx.x;
  red[tid] = v;
  __syncthreads();
  for (int off = 128; off > 0; off >>= 1) {
    if (tid < off) red[tid] += red[tid + off];
    __syncthreads();
  }
  float r = red[0];
  __syncthreads();
  return r;
}

template <int NN>
__device__ void jacobi_sym(float A[NN][NN], float V[NN][NN], int sweeps) {
  for (int i = 0; i < NN; i++)
    for (int j = 0; j < NN; j++) V[i][j] = (i == j) ? 1.0f : 0.0f;
  for (int sw = 0; sw < sweeps; sw++) {
    for (int p = 0; p < NN - 1; p++) {
      for (int q = p + 1; q < NN; q++) {
        float apq = A[p][q];
        if (fabsf(apq) < 1e-25f) continue;
        float phi = 0.5f * atan2f(2.0f * apq, A[q][q] - A[p][p]);
        float c = cosf(phi), s = sinf(phi);
        for (int k = 0; k < NN; k++) {
          float akp = A[k][p], akq = A[k][q];
          A[k][p] = c * akp - s * akq;
          A[k][q] = s * akp + c * akq;
        }
        for (int k = 0; k < NN; k++) {
          float apk = A[p][k], aqk = A[q][k];
          A[p][k] = c * apk - s * aqk;
          A[q][k] = s * apk + c * aqk;
        }
        for (int k = 0; k < NN; k++) {
          float vkp = V[k][p], vkq = V[k][q];
          V[k][p] = c * vkp - s * vkq;
          V[k][q] = s * vkp + c * vkq;
        }
      }
    }
  }
}

// From 9x9 Gram G (destroyed) + normalization params, produce final 3x3
// out = T2^T * rank2(F) * T1, and optionally the unit 9-vector f.
__device__ void solve_from_gram(float G[9][9], float s1, float cx1, float cy1,
                                float s2, float cx2, float cy2,
                                float* out9, float* fvec) {
  float V[9][9];
  jacobi_sym<9>(G, V, 12);
  int im = 0;
  float mv = G[0][0];
  for (int i = 1; i < 9; i++)
    if (G[i][i] < mv) { mv = G[i][i]; im = i; }
  float f[9];
  for (int i = 0; i < 9; i++) f[i] = V[i][im];
  if (fvec) for (int i = 0; i < 9; i++) fvec[i] = f[i];
  float F[3][3] = {{f[0], f[1], f[2]}, {f[3], f[4], f[5]}, {f[6], f[7], f[8]}};
  // rank-2 projection: Fo = F - (F v3) v3^T, v3 = min eigenvector of F^T F
  float A3[3][3];
  for (int i = 0; i < 3; i++)
    for (int j = 0; j < 3; j++) {
      float s = 0.0f;
      for (int k = 0; k < 3; k++) s += F[k][i] * F[k][j];
      A3[i][j] = s;
    }
  float V3[3][3];
  jacobi_sym<3>(A3, V3, 8);
  int i3 = 0;
  float m3 = A3[0][0];
  for (int i = 1; i < 3; i++)
    if (A3[i][i] < m3) { m3 = A3[i][i]; i3 = i; }
  float v3[3] = {V3[0][i3], V3[1][i3], V3[2][i3]};
  float Fv3[3];
  for (int i = 0; i < 3; i++)
    Fv3[i] = F[i][0] * v3[0] + F[i][1] * v3[1] + F[i][2] * v3[2];
  float Fo[3][3];
  for (int i = 0; i < 3; i++)
    for (int j = 0; j < 3; j++) Fo[i][j] = F[i][j] - Fv3[i] * v3[j];
  float T1[3][3] = {{s1, 0.0f, -cx1 * s1}, {0.0f, s1, -cy1 * s1}, {0.0f, 0.0f, 1.0f}};
  float T2[3][3] = {{s2, 0.0f, -cx2 * s2}, {0.0f, s2, -cy2 * s2}, {0.0f, 0.0f, 1.0f}};
  float G1[3][3];
  for (int i = 0; i < 3; i++)
    for (int j = 0; j < 3; j++) {
      float s = 0.0f;
      for (int k = 0; k < 3; k++) s += T2[k][i] * Fo[k][j];
      G1[i][j] = s;
    }
  for (int i = 0; i < 3; i++)
    for (int j = 0; j < 3; j++) {
      float s = 0.0f;
      for (int k = 0; k < 3; k++) s += G1[i][k] * T1[k][j];
      out9[i * 3 + j] = s;
    }
}

// scalar weighted 8-point fit on <=20 points (per-thread)
__device__ void small_wsvd(const float* lx1, const float* ly1, const float* lx2,
                           const float* ly2, const float* lw, int n,
                           float* out9, float* res) {
  float cx1 = 0, cy1 = 0, cx2 = 0, cy2 = 0;
  for (int i = 0; i < n; i++) { cx1 += lx1[i]; cy1 += ly1[i]; cx2 += lx2[i]; cy2 += ly2[i]; }
  float inv = 1.0f / (float)n;
  cx1 *= inv; cy1 *= inv; cx2 *= inv; cy2 *= inv;
  float sd1 = 0, sd2 = 0;
  for (int i = 0; i < n; i++) {
    float dx = lx1[i] - cx1, dy = ly1[i] - cy1;
    sd1 += __builtin_amdgcn_sqrtf(dx * dx + dy * dy);
    float ex = lx2[i] - cx2, ey = ly2[i] - cy2;
    sd2 += __builtin_amdgcn_sqrtf(ex * ex + ey * ey);
  }
  float s1 = 1.4142f * __builtin_amdgcn_rcpf(sd1 * inv + 1e-20f);
  float s2 = 1.4142f * __builtin_amdgcn_rcpf(sd2 * inv + 1e-20f);
  float G[9][9];
  for (int i = 0; i < 9; i++)
    for (int j = 0; j < 9; j++) G[i][j] = 0.0f;
  for (int p = 0; p < n; p++) {
    float a1 = s1 * (lx1[p] - cx1), b1 = s1 * (ly1[p] - cy1);
    float a2 = s2 * (lx2[p] - cx2), b2 = s2 * (ly2[p] - cy2);
    float q1 = a1 * a1 + b1 * b1 + 1.0f;
    float q2 = a2 * a2 + b2 * b2 + 1.0f;
    float sc = lw[p] * __builtin_amdgcn_rsqf(q1 * q2);
    float r[9] = {a2 * a1, a2 * b1, a2, b2 * a1, b2 * b1, b2, a1, b1, 1.0f};
    for (int i = 0; i < 9; i++) {
      float ri = r[i] * sc;
      for (int j = 0; j < 9; j++) G[i][j] += ri * (r[j] * sc);
    }
  }
  float f[9];
  solve_from_gram(G, s1, cx1, cy1, s2, cx2, cy2, out9, f);
  if (res) {
    float F[3][3] = {{f[0], f[1], f[2]}, {f[3], f[4], f[5]}, {f[6], f[7], f[8]}};
    for (int p = 0; p < n; p++) {
      float a1 = s1 * (lx1[p] - cx1), b1 = s1 * (ly1[p] - cy1);
      float a2 = s2 * (lx2[p] - cx2), b2 = s2 * (ly2[p] - cy2);
      float q1 = a1 * a1 + b1 * b1 + 1.0f;
      float q2 = a2 * a2 + b2 * b2 + 1.0f;
      float t0 = F[0][0] * a1 + F[0][1] * b1 + F[0][2];
      float t1 = F[1][0] * a1 + F[1][1] * b1 + F[1][2];
      float t2 = F[2][0] * a1 + F[2][1] * b1 + F[2][2];
      float dot = a2 * t0 + b2 * t1 + t2;
      res[p] = lw[p] * dot * __builtin_amdgcn_rsqf(q1 * q2);
    }
  }
}

// ---------------------------------------------------------------------------
// kernel 1: main fit — one workgroup per batch.
// Each stage: 256 threads normalize 256 points once and store the full
// 9-element Kronecker row (stride 12, b128-aligned) into LDS; then each of
// the 8 waves runs 8 unrolled V_WMMA_F32_16X16X4_F32 steps whose operands are
// single ds_load_b32 reads (no per-step selection branches).  Stage count is a
// uniform sgpr loop so EXEC is all-ones around every WMMA.  The next stage's
// points are prefetched (global_prefetch_b8) during staging.
// ---------------------------------------------------------------------------
__global__ __launch_bounds__(256) void fit_main_kernel(
    const float* __restrict__ pts1, const float* __restrict__ pts2,
    const float* __restrict__ weights, int N, float* __restrict__ out,
    float* __restrict__ params) {
  int b = blockIdx.x;
  const float* P1 = pts1 + (size_t)b * N * 3;
  const float* P2 = pts2 + (size_t)b * N * 3;
  const float* W = weights + (size_t)b * N;
  __shared__ float red[256];
  __shared__ float gram[256];
  __shared__ float Srow[256 * 12 + 16];   // padding covers mi=12..15 reads
  int tid = threadIdx.x;

  // pass 1: centroids (data stays L2-resident across passes)
  float sx1 = 0, sy1 = 0, sx2 = 0, sy2 = 0;
  for (int n = tid; n < N; n += 256) {
    sx1 += P1[n * 3 + 0]; sy1 += P1[n * 3 + 1];
    sx2 += P2[n * 3 + 0]; sy2 += P2[n * 3 + 1];
  }
  float inv = 1.0f / (float)N;
  float cx1 = block_reduce_sum(sx1, red) * inv;
  float cy1 = block_reduce_sum(sy1, red) * inv;
  float cx2 = block_reduce_sum(sx2, red) * inv;
  float cy2 = block_reduce_sum(sy2, red) * inv;

  // pass 2: mean distances
  float d1 = 0, d2 = 0;
  for (int n = tid; n < N; n += 256) {
    float dx = P1[n * 3 + 0] - cx1, dy = P1[n * 3 + 1] - cy1;
    d1 += __builtin_amdgcn_sqrtf(dx * dx + dy * dy);
    float ex = P2[n * 3 + 0] - cx2, ey = P2[n * 3 + 1] - cy2;
    d2 += __builtin_amdgcn_sqrtf(ex * ex + ey * ey);
  }
  float s1 = 1.4142f * __builtin_amdgcn_rcpf(block_reduce_sum(d1, red) * inv + 1e-20f);
  float s2 = 1.4142f * __builtin_amdgcn_rcpf(block_reduce_sum(d2, red) * inv + 1e-20f);

  // pass 3: Gram via WMMA.
  gram[tid] = 0.0f;
  int lane = tid & 31, wave = tid >> 5;
  int mi = lane & 15;          // row (=col) index produced by this lane
  int kg = (lane >> 4) * 2;    // K sub-offset per f32 16x4-A / 4x16-B layout
  v8f acc = {};
  int nStages = (N + 255) >> 8;            // uniform sgpr trip count
  for (int st = 0; st < nStages; ++st) {
    int k = (st << 8) + tid;
    // normalize this point once and expand the full weighted unit row
    float r0 = 0, r1 = 0, r2 = 0, r3 = 0, r4 = 0, r5 = 0, r6 = 0, r7 = 0, r8 = 0;
    if (k < N) {
      float x1 = P1[k * 3 + 0], y1 = P1[k * 3 + 1];
      float x2 = P2[k * 3 + 0], y2 = P2[k * 3 + 1];
      float w = W[k];
      float a1 = s1 * (x1 - cx1), b1 = s1 * (y1 - cy1);
      float a2 = s2 * (x2 - cx2), b2 = s2 * (y2 - cy2);
      float q1 = a1 * a1 + b1 * b1 + 1.0f;
      float q2 = a2 * a2 + b2 * b2 + 1.0f;
      float sc = w * __builtin_amdgcn_rsqf(q1 * q2);   // one v_rsq, no divide
      float c0 = a2 * sc, c1 = b2 * sc, c2 = sc;
      r0 = c0 * a1; r1 = c0 * b1; r2 = c0;
      r3 = c1 * a1; r4 = c1 * b1; r5 = c1;
      r6 = c2 * a1; r7 = c2 * b1; r8 = c2;
    }
    int kn = k + 256;                       // prefetch next stage while staging
    if (kn < N) {
      __builtin_prefetch(&P1[kn * 3], 0, 1);
      __builtin_prefetch(&P2[kn * 3], 0, 1);
      __builtin_prefetch(&W[kn], 0, 1);
    }
    float4* sr = (float4*)&Srow[tid * 12];
    sr[0] = make_float4(r0, r1, r2, r3);
    sr[1] = make_float4(r4, r5, r6, r7);
    Srow[tid * 12 + 8] = r8;
    __syncthreads();
    int qbase = wave * 32 + kg;            // this wave's 32 points of the stage
    #pragma unroll
    for (int j = 0; j < 8; ++j) {
      int pA = qbase + j * 4;
      float eA = Srow[pA * 12 + mi];       // single LDS word per operand
      float eB = Srow[(pA + 1) * 12 + mi];
      v2f a;
      a.x = (mi < 9) ? eA : 0.0f;
      a.y = (mi < 9) ? eB : 0.0f;
      // symmetric product: f32 16x4-A and 4x16-B lane layouts coincide -> B=A
      acc = __builtin_amdgcn_wmma_f32_16x16x4_f32(false, a, false, a, (short)0,
                                                  acc, false, false);
    }
    __syncthreads();
  }
  // merge the 8 per-wave partial C tiles
  int M0 = (lane >> 4) * 8;    // C layout: vgpr v, lane<16 -> M=v, else M=v+8
  #pragma unroll
  for (int v = 0; v < 8; v++) atomicAdd(&gram[(M0 + v) * 16 + mi], acc[v]);
  __syncthreads();

  if (tid == 0) {
    float G[9][9];
    for (int i = 0; i < 9; i++)
      for (int j = 0; j < 9; j++) G[i][j] = gram[i * 16 + j];
    float f[9];
    solve_from_gram(G, s1, cx1, cy1, s2, cx2, cy2, out + (size_t)b * 9, f);
    float* prm = params + (size_t)b * 16;
    prm[0] = s1; prm[1] = cx1; prm[2] = cy1;
    prm[3] = s2; prm[4] = cx2; prm[5] = cy2;
    for (int i = 0; i < 9; i++) prm[6 + i] = f[i];
  }
}

// ---------------------------------------------------------------------------
// kernel 2: residual[b,n] = w * p2h^T F p1h * rsqrt(q1*q2)
// ---------------------------------------------------------------------------
__global__ __launch_bounds__(256) void residual_kernel(
    const float* __restrict__ pts1, const float* __restrict__ pts2,
    const float* __restrict__ weights, const float* __restrict__ params,
    float* __restrict__ resid, int N, int total) {
  int i = blockIdx.x * blockDim.x + threadIdx.x;
  if (i >= total) return;
  int b = i / N, n = i - b * N;
  const float* prm = params + (size_t)b * 16;
  float s1 = prm[0], cx1 = prm[1], cy1 = prm[2];
  float s2 = prm[3], cx2 = prm[4], cy2 = prm[5];
  const float* f = prm + 6;
  const float* P1 = pts1 + ((size_t)b * N + n) * 3;
  const float* P2 = pts2 + ((size_t)b * N + n) * 3;
  float w = weights[(size_t)b * N + n];
  float a1 = s1 * (P1[0] - cx1), b1 = s1 * (P1[1] - cy1);
  float a2 = s2 * (P2[0] - cx2), b2 = s2 * (P2[1] - cy2);
  float q1 = a1 * a1 + b1 * b1 + 1.0f;
  float q2 = a2 * a2 + b2 * b2 + 1.0f;
  float t0 = f[0] * a1 + f[1] * b1 + f[2];
  float t1 = f[3] * a1 + f[4] * b1 + f[5];
  float t2 = f[6] * a1 + f[7] * b1 + f[8];
  float dot = a2 * t0 + b2 * t1 + t2;
  resid[i] = w * dot * __builtin_amdgcn_rsqf(q1 * q2);
}

// ---------------------------------------------------------------------------
// kernel 3: per-batch top-20 (first-index tie break, valid range only)
// ---------------------------------------------------------------------------
__global__ __launch_bounds__(256) void topk_kernel(
    const float* __restrict__ weights, const int* __restrict__ mgu,
    int* __restrict__ topidx, int N) {
  int b = blockIdx.x;
  const float* W = weights + (size_t)b * N;
  int M = mgu[b];
  __shared__ float sv[256];
  __shared__ int si[256];
  __shared__ int found[TOPK_];
  for (int t = 0; t < TOPK_; t++) {
    float best = -3.4e38f;
    int bi = 0x7fffffff;
    for (int n = threadIdx.x; n < M; n += 256) {
      bool skip = false;
      for (int u = 0; u < t; u++)
        if (found[u] == n) skip = true;
      if (skip) continue;
      float w = W[n];
      if (w > best || (w == best && n < bi)) { best = w; bi = n; }
    }
    sv[threadIdx.x] = best;
    si[threadIdx.x] = bi;
    __syncthreads();
    for (int off = 128; off > 0; off >>= 1) {
      if (threadIdx.x < off) {
        float wo = sv[threadIdx.x + off];
        int io = si[threadIdx.x + off];
        if (wo > sv[threadIdx.x] || (wo == sv[threadIdx.x] && io < si[threadIdx.x])) {
          sv[threadIdx.x] = wo;
          si[threadIdx.x] = io;
        }
      }
      __syncthreads();
    }
    if (threadIdx.x == 0) found[t] = (si[0] == 0x7fffffff) ? 0 : si[0];
    __syncthreads();
  }
  if (threadIdx.x < TOPK_) topidx[(size_t)b * TOPK_ + threadIdx.x] = found[threadIdx.x];
}

// ---------------------------------------------------------------------------
// kernel 4: top-20 re-fit (one thread per batch)
// ---------------------------------------------------------------------------
__global__ __launch_bounds__(64) void outk_kernel(
    const float* __restrict__ pts1, const float* __restrict__ pts2,
    const float* __restrict__ weights, const int* __restrict__ topidx,
    float* __restrict__ outK, float* __restrict__ resK, int B, int N) {
  int b = blockIdx.x * blockDim.x + threadIdx.x;
  if (b >= B) return;
  float lx1[TOPK_], ly1[TOPK_], lx2[TOPK_], ly2[TOPK_], lw[TOPK_];
  for (int t = 0; t < TOPK_; t++) {
    int n = topidx[(size_t)b * TOPK_ + t];
    lx1[t] = pts1[((size_t)b * N + n) * 3 + 0];
    ly1[t] = pts1[((size_t)b * N + n) * 3 + 1];
    lx2[t] = pts2[((size_t)b * N + n) * 3 + 0];
    ly2[t] = pts2[((size_t)b * N + n) * 3 + 1];
    lw[t] = weights[(size_t)b * N + n];
  }
  small_wsvd(lx1, ly1, lx2, ly2, lw, TOPK_, outK + (size_t)b * 9,
             resK + (size_t)b * TOPK_);
}

// ---------------------------------------------------------------------------
// kernel 5: per-batch inclusive prefix sum of masked weights (for sampling)
// ---------------------------------------------------------------------------
__global__ __launch_bounds__(256) void scan_kernel(
    const float* __restrict__ weights, const int* __restrict__ mgu,
    float* __restrict__ prefix, int N) {
  int b = blockIdx.x;
  const float* W = weights + (size_t)b * N;
  int M = mgu[b];
  __shared__ float buf[256];
  __shared__ float carry;
  int tid = threadIdx.x;
  if (tid == 0) carry = 0.0f;
  __syncthreads();
  for (int base = 0; base < N; base += 256) {
    int n = base + tid;
    float v = (n < N && n < M) ? W[n] : 0.0f;
    buf[tid] = v;
    __syncthreads();
    for (int off = 1; off < 256; off <<= 1) {
      float t = (tid >= off) ? buf[tid - off] : 0.0f;
      __syncthreads();
      buf[tid] += t;
      __syncthreads();
    }
    float incl = buf[tid];
    float c = carry;
    if (n < N) prefix[(size_t)b * N + n] = incl + c;
    __syncthreads();
    if (tid == 255) carry = c + incl;
    __syncthreads();
  }
}

// ---------------------------------------------------------------------------
// kernel 6: deterministic categorical sampling via inverse CDF
// ---------------------------------------------------------------------------
__global__ __launch_bounds__(256) void sample_kernel(
    const float* __restrict__ prefix, const int* __restrict__ mgu,
    int* __restrict__ sel, int B, int N) {
  int i = blockIdx.x * blockDim.x + threadIdx.x;
  int total = B * SELECTS_ * TOPK_;
  if (i >= total) return;
  int b = i / (SELECTS_ * TOPK_);
  const float* pf = prefix + (size_t)b * N;
  int M = mgu[b];
  float tot = pf[M - 1];
  uint32_t h = (uint32_t)i * 0x9E3779B9u + 0x85EBCA6Bu;
  h ^= h >> 16; h *= 0x7FEB352Du; h ^= h >> 15; h *= 0x846CA68Bu; h ^= h >> 16;
  float u = (float)(h >> 8) * (1.0f / 16777216.0f);
  float target = u * tot;
  int lo = 0, hi = M - 1;
  while (lo < hi) {
    int mid = (lo + hi) >> 1;
    if (pf[mid] >= target) hi = mid; else lo = mid + 1;
  }
  sel[i] = lo;
}

// ---------------------------------------------------------------------------
// kernel 7: accu[b,s] = prod(ws*1000) / (sum_s prod + 1e-10)
// ---------------------------------------------------------------------------
__global__ __launch_bounds__(128) void accu_kernel(
    const float* __restrict__ weights, const int* __restrict__ sel,
    float* __restrict__ accu, int N) {
  int b = blockIdx.x, s = threadIdx.x;
  __shared__ float red[128];
  float prod = 0.0f;
  if (s < SELECTS_) {
    prod = 1.0f;
    const int* sp = sel + ((size_t)b * SELECTS_ + s) * TOPK_;
    for (int t = 0; t < TOPK_; t++)
      prod *= weights[(size_t)b * N + sp[t]] * 1000.0f;
  }
  red[s] = (s < SELECTS_) ? prod : 0.0f;
  __syncthreads();
  for (int off = 64; off > 0; off >>= 1) {
    if (s < off) red[s] += red[s + off];
    __syncthreads();
  }
  float tot = red[0];
  if (s < SELECTS_) accu[(size_t)b * SELECTS_ + s] = prod / (tot + 1e-10f);
}

// ---------------------------------------------------------------------------
// kernel 8: 12800 sampled re-fits (one thread per (b,s))
// ---------------------------------------------------------------------------
__global__ __launch_bounds__(128) void selects_kernel(
    const float* __restrict__ pts1, const float* __restrict__ pts2,
    const float* __restrict__ weights, const int* __restrict__ sel,
    float* __restrict__ outS, int B, int N) {
  int i = blockIdx.x * blockDim.x + threadIdx.x;
  if (i >= B * SELECTS_) return;
  int b = i / SELECTS_;
  float lx1[TOPK_], ly1[TOPK_], lx2[TOPK_], ly2[TOPK_], lw[TOPK_];
  const int* sp = sel + (size_t)i * TOPK_;
  for (int t = 0; t < TOPK_; t++) {
    int n = sp[t];
    lx1[t] = pts1[((size_t)b * N + n) * 3 + 0];
    ly1[t] = pts1[((size_t)b * N + n) * 3 + 1];
    lx2[t] = pts2[((size_t)b * N + n) * 3 + 0];
    ly2[t] = pts2[((size_t)b * N + n) * 3 + 1];
    lw[t] = weights[(size_t)b * N + n];
  }
  small_wsvd(lx1, ly1, lx2, ly2, lw, TOPK_, outS + (size_t)i * 9, nullptr);
}

// ---------------------------------------------------------------------------
extern "C" void kernel_launch(void* const* d_in, const int* in_sizes, int n_in,
                              void* d_out, int out_size, void* d_ws,
                              size_t ws_size, hipStream_t stream) {
  const float* pts1 = (const float*)d_in[0];
  const float* pts2 = (const float*)d_in[1];
  const float* weights = (const float*)d_in[2];
  const int* mgu = (const int*)d_in[3];
  int B = in_sizes[3];
  int N = in_sizes[2] / B;   // weights is (B,1,N)

  // output layout (concatenated flat in return order)
  float* out = (float*)d_out;
  float* resid = out + (size_t)B * 9;
  float* outK = resid + (size_t)B * N;
  float* resK = outK + (size_t)B * 9;
  float* outS = resK + (size_t)B * TOPK_;
  float* accu = outS + (size_t)B * SELECTS_ * 9;

  // workspace layout
  char* wsb = (char*)d_ws;
  size_t off = 0;
  auto al = [](size_t x) { return (x + 255) & ~(size_t)255; };
  float* params = (float*)(wsb + off);
  off = al(off + (size_t)B * 16 * sizeof(float));
  int* topidx = (int*)(wsb + off);
  off = al(off + (size_t)B * TOPK_ * sizeof(int));
  float* prefix = (float*)(wsb + off);
  off = al(off + (size_t)B * N * sizeof(float));
  int* sel = (int*)(wsb + off);
  off = al(off + (size_t)B * SELECTS_ * TOPK_ * sizeof(int));
  (void)ws_size;

  fit_main_kernel<<<B, 256, 0, stream>>>(pts1, pts2, weights, N, out, params);
  int totBN = B * N;
  residual_kernel<<<(totBN + 255) / 256, 256, 0, stream>>>(
      pts1, pts2, weights, params, resid, N, totBN);
  topk_kernel<<<B, 256, 0, stream>>>(weights, mgu, topidx, N);
  outk_kernel<<<(B + 63) / 64, 64, 0, stream>>>(pts1, pts2, weights, topidx,
                                                outK, resK, B, N);
  scan_kernel<<<B, 256, 0, stream>>>(weights, mgu, prefix, N);
  int totSel = B * SELECTS_ * TOPK_;
  sample_kernel<<<(totSel + 255) / 256, 256, 0, stream>>>(prefix, mgu, sel, B, N);
  accu_kernel<<<B, 128, 0, stream>>>(weights, sel, accu, N);
  selects_kernel<<<(B * SELECTS_ + 127) / 128, 128, 0, stream>>>(
      pts1, pts2, weights, sel, outS, B, N);
}